// Attention_80925773791340
// MI455X (gfx1250) — compile-verified
//
#include <hip/hip_runtime.h>

// ---------------------------------------------------------------------------
// Attention with MLP relative-position bias, fp32, CDNA5 (gfx1250, wave32).
// All GEMM-like math on V_WMMA_F32_16X16X4_F32 (exact fp32 vs reference).
// Flash-style attention in transposed orientation so softmax state is
// per-lane.  Bias MLP folded into a 63x63x12 lookup table.
// ---------------------------------------------------------------------------

typedef float v2f __attribute__((ext_vector_type(2)));
typedef float v8f __attribute__((ext_vector_type(8)));

#define DIMC   768
#define NHEAD  12
#define HDIM   64
#define SEQ    1024
#define BATCH  8
#define RPBH   64
#define TBL    63          // 2*32-1 distinct relative offsets per axis
#define SCALE  0.125f      // 64^-0.5

static __device__ __forceinline__ v8f wmma_f32(v2f a, v2f b, v8f c) {
  // 8 args: (neg_a, A, neg_b, B, c_mod, C, reuse_a, reuse_b)
  return __builtin_amdgcn_wmma_f32_16x16x4_f32(false, a, false, b,
                                               (short)0, c, false, false);
}

// ---------------------------------------------------------------------------
// Kernel 1/4: C[M,Nn] = A[M,K] * W[Nn,K]^T (+ bias).
// Wave -> 32x64 tile (2 A-frags x 4 B-frags -> 8 WMMAs per k-step, 6 loads).
// Block of 4 waves (2x2) -> 64x128 tile.  Fragments loaded as b64 per lane
// straight from L2-resident operands.
// ---------------------------------------------------------------------------
__global__ void gemm_wmma(const float* __restrict__ A, const float* __restrict__ W,
                          const float* __restrict__ bias, float* __restrict__ C,
                          int M, int Nn, int K) {
  const int lane  = threadIdx.x & 31;
  const int wid   = threadIdx.x >> 5;              // 0..3
  const int r0    = blockIdx.x * 64  + (wid & 1) * 32;
  const int c0    = blockIdx.y * 128 + (wid >> 1) * 64;
  const int lrow  = lane & 15;
  const int khalf = (lane >> 4) << 1;              // 0 or 2 (K sub-offset)

  v8f acc[2][4];
#pragma unroll
  for (int i = 0; i < 2; i++)
#pragma unroll
    for (int j = 0; j < 4; j++) acc[i][j] = (v8f){};

  const float* ap[2];
  const float* bp[4];
#pragma unroll
  for (int i = 0; i < 2; i++)
    ap[i] = A + (size_t)(r0 + i * 16 + lrow) * K + khalf;
#pragma unroll
  for (int j = 0; j < 4; j++)
    bp[j] = W + (size_t)(c0 + j * 16 + lrow) * K + khalf;

  for (int k0 = 0; k0 < K; k0 += 4) {
    v2f af[2], bf[4];
#pragma unroll
    for (int i = 0; i < 2; i++) af[i] = *(const v2f*)(ap[i] + k0);
#pragma unroll
    for (int j = 0; j < 4; j++) bf[j] = *(const v2f*)(bp[j] + k0);
#pragma unroll
    for (int i = 0; i < 2; i++)
#pragma unroll
      for (int j = 0; j < 4; j++)
        acc[i][j] = wmma_f32(af[i], bf[j], acc[i][j]);
  }

  const int rhalf = (lane >> 4) << 3;              // 0 or 8 (row offset in D)
#pragma unroll
  for (int i = 0; i < 2; i++) {
#pragma unroll
    for (int j = 0; j < 4; j++) {
      int col  = c0 + j * 16 + lrow;
      float bv = bias ? bias[col] : 0.0f;
#pragma unroll
      for (int v = 0; v < 8; v++) {
        int row = r0 + i * 16 + v + rhalf;
        C[(size_t)row * Nn + col] = acc[i][j][v] + bv;
      }
    }
  }
}

// ---------------------------------------------------------------------------
// Kernel 2: collapse the bias MLP onto the 63x63 distinct (rel_x, rel_y)
// grid: table[h][ry+31][rx+31].  ~4K threads, trivial cost.
// ---------------------------------------------------------------------------
__global__ void rpb_table_kernel(const float* __restrict__ w1, const float* __restrict__ b1,
                                 const float* __restrict__ w2, const float* __restrict__ b2,
                                 float* __restrict__ tabl) {
  int id = blockIdx.x * blockDim.x + threadIdx.x;
  if (id >= TBL * TBL) return;
  float rx = (float)(id % TBL - 31);
  float ry = (float)(id / TBL - 31);
  float acc[NHEAD];
#pragma unroll
  for (int t = 0; t < NHEAD; t++) acc[t] = b2[t];
  for (int j = 0; j < RPBH; j++) {
    float hv = fmaxf(w1[2 * j] * rx + w1[2 * j + 1] * ry + b1[j], 0.0f);
#pragma unroll
    for (int t = 0; t < NHEAD; t++) acc[t] += hv * w2[t * RPBH + j];
  }
#pragma unroll
  for (int t = 0; t < NHEAD; t++) tabl[t * TBL * TBL + id] = acc[t];
}

// ---------------------------------------------------------------------------
// Kernel 3: flash attention, transposed orientation.
//   S^T[m,r] = sum_d K[m,d] Q[r,d]      (16 WMMAs, K=64)
//   O^T[d,r] = sum_m V[m,d] P^T[m,r]    (16 WMMAs per key tile)
// Each lane owns query row r = lane&15 (duplicated across halves), so the
// online-softmax max/sum needs only one shfl_xor(16) and the accumulator
// rescale is per-lane.  P^T (C/D layout) -> B-fragments via 8 half-shuffles.
// One wave per (b, h, 16-row tile): 8*12*64 = 6144 waves.
// ---------------------------------------------------------------------------
__global__ void attn_flash(const float* __restrict__ qkv,
                           const float* __restrict__ tabl,
                           float* __restrict__ obuf) {
  const int lane = threadIdx.x & 31;
  const int gw   = blockIdx.x * (blockDim.x >> 5) + (threadIdx.x >> 5);
  const int qt   = gw & 63;                        // SEQ/16 tiles
  const int h    = (gw >> 6) % NHEAD;
  const int b    = gw / (64 * NHEAD);
  const int lr   = lane & 15;
  const int hi   = lane >> 4;                      // half select
  const int qrow0 = qt * 16;
  const int strd  = 3 * DIMC;                      // 2304

  const size_t base  = (size_t)b * SEQ * strd;
  const float* qbase = qkv + base + h * HDIM;
  const float* kbase = qkv + base + DIMC + h * HDIM;
  const float* vbase = qkv + base + 2 * DIMC + h * HDIM;

  // Preload Q^T B-fragments: qf[s] holds Q[r, 4s+2*hi .. +1] (loop invariant).
  v2f qf[16];
  {
    const float* qp = qbase + (size_t)(qrow0 + lr) * strd + 2 * hi;
#pragma unroll
    for (int s = 0; s < 16; s++) qf[s] = *(const v2f*)(qp + s * 4);
  }

  const int n  = qrow0 + lr;                       // this lane's query row
  const int nx = n & 31, ny = n >> 5;
  const float* tb = tabl + h * (TBL * TBL);

  v8f o0 = {}, o1 = {}, o2 = {}, o3 = {};          // O^T, d = t*16 + j(+8)
  float mrow = -1e30f, lsum = 0.0f;

  for (int m0 = 0; m0 < SEQ; m0 += 16) {
    // ---- S^T tile: A from K rows, B = preloaded Q^T fragments -------------
    v8f s = {};
    const float* kp = kbase + (size_t)(m0 + lr) * strd + 2 * hi;
#pragma unroll
    for (int t = 0; t < 16; t++) {
      v2f kf = *(const v2f*)(kp + t * 4);
      s = wmma_f32(kf, qf[t], s);
    }

    // ---- scale + bias gather + online softmax ----------------------------
    float p[8];
    float tmax = -1e30f;
#pragma unroll
    for (int j = 0; j < 8; j++) {
      int mk = m0 + j + (hi ? 8 : 0);              // key index of s[j]
      int mx = mk & 31, my = mk >> 5;
      float v = s[j] * SCALE + tb[(ny - my + 31) * TBL + (nx - mx + 31)];
      p[j] = v;
      tmax = fmaxf(tmax, v);
    }
    tmax = fmaxf(tmax, __shfl_xor(tmax, 16, 32));  // combine halves of row
    float mnew  = fmaxf(mrow, tmax);
    float alpha = __expf(mrow - mnew);
    float tsum = 0.0f;
#pragma unroll
    for (int j = 0; j < 8; j++) {
      p[j] = __expf(p[j] - mnew);
      tsum += p[j];
    }
    tsum += __shfl_xor(tsum, 16, 32);
    lsum = lsum * alpha + tsum;
    mrow = mnew;
#pragma unroll
    for (int j = 0; j < 8; j++) {                  // per-lane rescale of O^T
      o0[j] *= alpha; o1[j] *= alpha; o2[j] *= alpha; o3[j] *= alpha;
    }

    // ---- build P^T B-fragments (C/D layout -> B layout) -------------------
    // B vgpr for k-step k0: lanes<16 need m=k0(,+1), lanes>=16 need m=k0+2(,+3).
    v2f bfv[4];
    const int srcLo = lr;            // pull from low half
    const int srcHi = lr + 16;       // pull from high half
#pragma unroll
    for (int ss = 0; ss < 2; ss++) { // k0 = 0, 4 : m in 0..7
      float x0 = p[4 * ss],     y0 = __shfl(p[4 * ss + 2], srcLo, 32);
      float x1 = p[4 * ss + 1], y1 = __shfl(p[4 * ss + 3], srcLo, 32);
      bfv[ss].x = hi ? y0 : x0;
      bfv[ss].y = hi ? y1 : x1;
    }
#pragma unroll
    for (int ss = 0; ss < 2; ss++) { // k0 = 8, 12 : m in 8..15
      float x0 = __shfl(p[4 * ss],     srcHi, 32), y0 = p[4 * ss + 2];
      float x1 = __shfl(p[4 * ss + 1], srcHi, 32), y1 = p[4 * ss + 3];
      bfv[2 + ss].x = hi ? y0 : x0;
      bfv[2 + ss].y = hi ? y1 : x1;
    }

    // ---- O^T += V^T x P^T -------------------------------------------------
#pragma unroll
    for (int t = 0; t < 4; t++) {
      int d = t * 16 + lr;
#pragma unroll
      for (int ss = 0; ss < 4; ss++) {
        int mk = m0 + ss * 4 + 2 * hi;
        v2f av;
        av.x = vbase[(size_t)mk * strd + d];
        av.y = vbase[(size_t)(mk + 1) * strd + d];
        v8f& accR = (t == 0 ? o0 : t == 1 ? o1 : t == 2 ? o2 : o3);
        accR = wmma_f32(av, bfv[ss], accR);
      }
    }
  }

  // ---- normalize & store as [B, N, H*D] for the projection GEMM ----------
  float inv = 1.0f / lsum;
  float* op = obuf + (size_t)(b * SEQ + qrow0 + lr) * DIMC + h * HDIM;
#pragma unroll
  for (int t = 0; t < 4; t++) {
    const v8f& accR = (t == 0 ? o0 : t == 1 ? o1 : t == 2 ? o2 : o3);
#pragma unroll
    for (int j = 0; j < 8; j++) {
      int d = t * 16 + j + (hi ? 8 : 0);
      op[d] = accR[j] * inv;
    }
  }
}

// ---------------------------------------------------------------------------
extern "C" void kernel_launch(void* const* d_in, const int* in_sizes, int n_in,
                              void* d_out, int out_size, void* d_ws, size_t ws_size,
                              hipStream_t stream) {
  const float* x      = (const float*)d_in[0];
  const float* qkv_w  = (const float*)d_in[1];
  const float* proj_w = (const float*)d_in[2];
  const float* proj_b = (const float*)d_in[3];
  const float* rpb_w1 = (const float*)d_in[4];
  const float* rpb_b1 = (const float*)d_in[5];
  const float* rpb_w2 = (const float*)d_in[6];
  const float* rpb_b2 = (const float*)d_in[7];

  float* qkv  = (float*)d_ws;                              // 8192*2304 f32
  float* tabl = qkv + (size_t)BATCH * SEQ * 3 * DIMC;      // 12*63*63 f32
  float* obuf = tabl + NHEAD * TBL * TBL;                  // 8192*768 f32
  float* out  = (float*)d_out;

  // bias table (tiny)
  rpb_table_kernel<<<(TBL * TBL + 127) / 128, 128, 0, stream>>>(
      rpb_w1, rpb_b1, rpb_w2, rpb_b2, tabl);

  // QKV projection: [8192,768] x [2304,768]^T -> [8192,2304]
  gemm_wmma<<<dim3(BATCH * SEQ / 64, 3 * DIMC / 128), 128, 0, stream>>>(
      x, qkv_w, nullptr, qkv, BATCH * SEQ, 3 * DIMC, DIMC);

  // fused biased-softmax attention: 6144 waves, 4 waves/block
  attn_flash<<<BATCH * NHEAD * (SEQ / 16) / 4, 128, 0, stream>>>(qkv, tabl, obuf);

  // output projection with bias: [8192,768] x [768,768]^T + b -> d_out
  gemm_wmma<<<dim3(BATCH * SEQ / 64, DIMC / 128), 128, 0, stream>>>(
      obuf, proj_w, proj_b, out, BATCH * SEQ, DIMC, DIMC);
}